// TrackAwareLoss_36369783062763
// MI455X (gfx1250) — compile-verified
//
#include <hip/hip_runtime.h>
#include <stdint.h>

// ---------------------------------------------------------------------------
// TrackAwareLoss for MI455X (gfx1250)
//
// Memory-bound streaming reduction: 256MB of fp32 logits read exactly once.
// Data movement uses the CDNA5 Tensor Data Mover (tensor_load_to_lds) with
// double-buffered 64KB tiles in LDS, synchronized via TENSORcnt + workgroup
// barriers. Compute: two-pass softmax with the row slice held in VGPRs
// (ds_load_b128 once), so all v_exp_f32 ops are independent (no serial
// online-rescale chain).
// ---------------------------------------------------------------------------

#define IGNORE_INDEX   (-100)
#define DUR_START      512
#define DUR_END        639
#define V_DIM          2048
#define ROWS_PER_TILE  8
#define TILE_ELEMS     (ROWS_PER_TILE * V_DIM)   /* 16384 elems = 64KB */
#define VEC_PER_LANE   (V_DIM / 128)             /* 16 float4 per lane   */
#define LOG2E          1.4426950408889634f
#define LN2            0.6931471805599453f

#if defined(__AMDGCN__) && __has_builtin(__builtin_amdgcn_tensor_load_to_lds)
#define USE_TDM 1
#else
#define USE_TDM 0
#endif

typedef __attribute__((ext_vector_type(4))) unsigned int v4u;
typedef __attribute__((ext_vector_type(8))) int          v8i;
typedef __attribute__((ext_vector_type(4))) int          v4i;

__global__ void ta_init_kernel(float* acc) {
    acc[0] = 0.0f;
    acc[1] = 0.0f;
}

#if USE_TDM
// Issue one TDM load of a contiguous 64KB tile (1-D tensor descriptor).
// D# layout per CDNA5 ISA ch.8 (group0: 128b, group1: 256b, group2/3: 128b).
// 6-arg builtin form: 4 descriptor groups + extra operand group (zeroed:
// all optional TDM features disabled) + cache-policy immediate.
__device__ __forceinline__ void tdm_load_tile(const float* gsrc, float* ldsDst) {
    uint64_t ga = (uint64_t)(uintptr_t)gsrc;
    uint32_t la = (uint32_t)(uintptr_t)ldsDst;   // low 32 bits = LDS byte offset

    v4u g0;
    g0.x = 1u;                                           // count=1, user desc
    g0.y = la;                                           // lds_addr
    g0.z = (uint32_t)ga;                                 // global_addr[31:0]
    g0.w = (uint32_t)((ga >> 32) & 0x01FFFFFFu)          // global_addr[56:32]
         | (2u << 30);                                   // type = 2 ("image")

    v8i g1;
    g1[0] = (int)(2u << 16);                             // data_size = 4 bytes
    g1[1] = (int)(((uint32_t)TILE_ELEMS & 0xFFFFu) << 16);   // tensor_dim0 lo16
    g1[2] = (int)((((uint32_t)TILE_ELEMS >> 16) & 0xFFFFu)   // tensor_dim0 hi16
                  | (1u << 16));                             // tensor_dim1 = 1 (lo16)
    g1[3] = (int)(((uint32_t)TILE_ELEMS & 0xFFFFu) << 16);   // tensor_dim1 hi=0, tile_dim0 = 16384
    g1[4] = 0;                                           // tile_dim1/2 unused
    g1[5] = (int)TILE_ELEMS;                             // tensor_dim0_stride lo32
    g1[6] = 0;
    g1[7] = 0;

    v4i g2;                                              // tensor_dim2=1, tensor_dim3=1
    g2[0] = 1; g2[1] = 1; g2[2] = 0; g2[3] = 0;
    v4i g3;                                              // tensor_dim4=1 (bits 79:48)
    g3[0] = 0; g3[1] = (int)(1u << 16); g3[2] = 0; g3[3] = 0;

    v8i g4;                                              // extra operand group: off
    g4[0] = 0; g4[1] = 0; g4[2] = 0; g4[3] = 0;
    g4[4] = 0; g4[5] = 0; g4[6] = 0; g4[7] = 0;

    __builtin_amdgcn_tensor_load_to_lds(g0, g1, g2, g3, g4, 0);
}
#endif

__global__ __launch_bounds__(256) void ta_loss_kernel(
    const float* __restrict__ logits,
    const int*   __restrict__ targets,
    const int*   __restrict__ track_ids,
    const float* __restrict__ sweights,
    float*       __restrict__ acc,
    int BS, int S, int numTiles, int tilesPerBlock)
{
#if USE_TDM
    __shared__ float buf[2][TILE_ELEMS];                 // 128 KB double buffer
#endif
    __shared__ float blkNum, blkDen;

    const int lane = threadIdx.x & 31;
    const int wave = threadIdx.x >> 5;

    if (threadIdx.x == 0) { blkNum = 0.0f; blkDen = 0.0f; }
    __syncthreads();

    const int tBegin = blockIdx.x * tilesPerBlock;
    const int tEnd   = (tBegin + tilesPerBlock < numTiles) ? (tBegin + tilesPerBlock)
                                                           : numTiles;

#if USE_TDM
    if (wave == 0 && tBegin < tEnd)
        tdm_load_tile(logits + (size_t)tBegin * TILE_ELEMS, &buf[0][0]);
#endif

    for (int t = tBegin; t < tEnd; ++t) {
        const int sel = (t - tBegin) & 1;

#if USE_TDM
        if (wave == 0) {
            if (t + 1 < tEnd) {
                tdm_load_tile(logits + (size_t)(t + 1) * TILE_ELEMS, &buf[sel ^ 1][0]);
                __builtin_amdgcn_s_wait_tensorcnt(1);    // tile t is resident
            } else {
                __builtin_amdgcn_s_wait_tensorcnt(0);
            }
        }
        __syncthreads();                                 // release readers
        const float* row = &buf[sel][wave * V_DIM];
#else
        const float* row = logits + (size_t)t * TILE_ELEMS + wave * V_DIM;
#endif
        const int r = t * ROWS_PER_TILE + wave;
        if (r < BS) {
            // ---- pass 0: pull this lane's 64 elements into VGPRs ----------
            // 16 independent ds_load_b128 (immediate offsets off one address)
            const float4* rv = (const float4*)row;
            float4 v[VEC_PER_LANE];
            #pragma unroll
            for (int it = 0; it < VEC_PER_LANE; ++it)
                v[it] = rv[it * 32 + lane];

            // ---- pass 1: row max (cheap v_max_f32 tree) -------------------
            float mx = -3.0e38f;
            #pragma unroll
            for (int it = 0; it < VEC_PER_LANE; ++it) {
                float m4 = fmaxf(fmaxf(v[it].x, v[it].y), fmaxf(v[it].z, v[it].w));
                mx = fmaxf(mx, m4);
            }
            #pragma unroll
            for (int off = 16; off > 0; off >>= 1)
                mx = fmaxf(mx, __shfl_xor(mx, off, 32));
            const float my = mx * LOG2E;                 // max in base-2 domain

            // ---- pass 2: sum of 2^(x*log2e - my), all exps independent ----
            float s = 0.0f;
            #pragma unroll
            for (int it = 0; it < VEC_PER_LANE; ++it) {
                s += __builtin_amdgcn_exp2f(fmaf(v[it].x, LOG2E, -my))
                   + __builtin_amdgcn_exp2f(fmaf(v[it].y, LOG2E, -my))
                   + __builtin_amdgcn_exp2f(fmaf(v[it].z, LOG2E, -my))
                   + __builtin_amdgcn_exp2f(fmaf(v[it].w, LOG2E, -my));
            }
            #pragma unroll
            for (int off = 16; off > 0; off >>= 1)
                s += __shfl_xor(s, off, 32);

            if (lane == 0) {
                int  tg    = targets[r];
                bool valid = (tg != IGNORE_INDEX);
                int  st    = valid ? tg : 0;
                float xt   = row[st];
                float lse  = LN2 * (my + __builtin_amdgcn_logf(s)); // v_log_f32
                float nll  = lse - xt;

                int  tk       = track_ids[r];
                bool shortDur = (tg >= DUR_START) && (tg <= DUR_END)
                              && ((tg - DUR_START) < 3);
                float w = 1.0f;
                if (shortDur) w = (tk == 0) ? 10.0f : ((tk == 1) ? 5.0f : 1.0f);

                float swt = sweights[r / S];
                if (valid) {
                    atomicAdd(&blkNum, nll * w * swt);
                    atomicAdd(&blkDen, 1.0f);
                }
            }
        }
        __syncthreads();                                 // buffer reuse fence
    }

    if (threadIdx.x == 0) {
        atomicAdd(&acc[0], blkNum);
        atomicAdd(&acc[1], blkDen);
    }
}

__global__ void ta_final_kernel(const float* acc, float* out) {
    out[0] = acc[0] / fmaxf(acc[1], 1.0f);
}

extern "C" void kernel_launch(void* const* d_in, const int* in_sizes, int n_in,
                              void* d_out, int out_size, void* d_ws, size_t ws_size,
                              hipStream_t stream) {
    const float* logits    = (const float*)d_in[0];
    const int*   targets   = (const int*)d_in[1];
    const int*   track_ids = (const int*)d_in[2];
    const float* sweights  = (const float*)d_in[3];

    const int BS = in_sizes[1];                  // B*S
    const int B  = in_sizes[3];
    const int S  = BS / B;
    // V is fixed at 2048 by the LDS tiling (in_sizes[0] / BS == 2048 here).

    const int numTiles = (BS + ROWS_PER_TILE - 1) / ROWS_PER_TILE;   // 4096
    int grid = 512;
    if (grid > numTiles) grid = numTiles;
    const int tilesPerBlock = (numTiles + grid - 1) / grid;          // 8

    float* acc = (float*)d_ws;                   // [num, den]

    ta_init_kernel<<<1, 1, 0, stream>>>(acc);
    ta_loss_kernel<<<grid, 256, 0, stream>>>(logits, targets, track_ids, sweights,
                                             acc, BS, S, numTiles, tilesPerBlock);
    ta_final_kernel<<<1, 1, 0, stream>>>(acc, (float*)d_out);
}